// GCN_89644557402315
// MI455X (gfx1250) — compile-verified
//
#include <hip/hip_runtime.h>
#include <hip/hip_bf16.h>

typedef float v2f __attribute__((ext_vector_type(2)));
typedef float v8f __attribute__((ext_vector_type(8)));

#define DIM 128
#define NT 17

// ---------- tiny GEMM: HW1[17x128] = emb[17x128] @ W1[128x128] ----------
__global__ __launch_bounds__(256) void k_hw1(const float* __restrict__ emb,
                                             const float* __restrict__ W1,
                                             float* __restrict__ HW1) {
    int t = blockIdx.x * 256 + threadIdx.x;
    if (t >= NT * DIM) return;
    int m = t >> 7, n = t & 127;
    float acc = 0.f;
    #pragma unroll 8
    for (int k = 0; k < DIM; ++k)
        acc = fmaf(emb[m * DIM + k], W1[k * DIM + n], acc);
    HW1[t] = acc;
}

// ---------- degree (with self-loop) ----------
__global__ __launch_bounds__(256) void k_deg_init(float* __restrict__ deg, int n) {
    int i = blockIdx.x * 256 + threadIdx.x;
    if (i < n) deg[i] = 1.0f;                     // self-loop contribution
}
__global__ __launch_bounds__(256) void k_deg_edges(const int* __restrict__ dst,
                                                   float* __restrict__ deg, int nE) {
    int e = blockIdx.x * 256 + threadIdx.x;
    if (e < nE) atomicAdd(&deg[dst[e]], 1.0f);
}
__global__ __launch_bounds__(256) void k_dinv(float* __restrict__ deg, int n) {
    int i = blockIdx.x * 256 + threadIdx.x;
    if (i < n) { float d = deg[i]; deg[i] = d > 0.f ? rsqrtf(d) : 0.f; }
}

// ---------- conv1 self-loop seed: agg1[i] = dinv[i]^2 * HW1[node_ids[i]] ----------
__global__ __launch_bounds__(256) void k_self1(const int* __restrict__ node_ids,
                                               const float* __restrict__ dinv,
                                               const float* __restrict__ HW1,
                                               float* __restrict__ agg, int n) {
    int t = blockIdx.x * 256 + threadIdx.x;
    int i = t >> 5;
    if (i >= n) return;
    int f = (t & 31) * 4;
    float w = dinv[i] * dinv[i];
    const float4 h = *(const float4*)(HW1 + (size_t)node_ids[i] * DIM + f);
    float4 o = make_float4(w * h.x, w * h.y, w * h.z, w * h.w);
    *(float4*)(agg + (size_t)i * DIM + f) = o;
}

// ---------- conv1 edge scatter: wave per edge, 4 f32 atomics per lane ----------
__global__ __launch_bounds__(256) void k_edge1(const int* __restrict__ src,
                                               const int* __restrict__ dst,
                                               const int* __restrict__ node_ids,
                                               const float* __restrict__ dinv,
                                               const float* __restrict__ HW1,
                                               float* __restrict__ agg, int nE) {
    int w = (blockIdx.x * 256 + threadIdx.x) >> 5;
    int lane = threadIdx.x & 31;
    if (w >= nE) return;
    int s = src[w], d = dst[w];
    float nrm = dinv[s] * dinv[d];
    const float* __restrict__ hrow = HW1 + (size_t)node_ids[s] * DIM;  // 8.5KB table, L1-hot
    float* __restrict__ orow = agg + (size_t)d * DIM;
    #pragma unroll
    for (int j = 0; j < 4; ++j) {
        int f = lane + 32 * j;
        atomicAdd(&orow[f], nrm * hrow[f]);
    }
}

// ---------- GEMM2 via WMMA f32 16x16x4: h2 = relu(agg1 + b1) @ W2 ----------
// 256 threads = 8 waves; block owns a 32-row strip; wave w owns cols [16w,16w+16)
// and computes TWO 16x16 row-tiles, sharing each B fragment across both WMMAs.
__global__ __launch_bounds__(256) void k_gemm2(const float* __restrict__ agg1,
                                               const float* __restrict__ b1,
                                               const float* __restrict__ W2,
                                               float* __restrict__ h2, int n) {
    __shared__ float As[32][DIM + 4];            // +4 pad: kills LDS bank conflicts
    int r0 = blockIdx.x * 32;
    int tid = threadIdx.x;
    for (int t = tid; t < 32 * DIM; t += 256) {
        int r = t >> 7, c = t & 127;
        float v = 0.f;
        if (r0 + r < n) {
            v = agg1[(size_t)(r0 + r) * DIM + c] + b1[c];
            v = fmaxf(v, 0.f);                   // fused ReLU
        }
        As[r][c] = v;
    }
    __syncthreads();

    int wave = tid >> 5;
    int lane = tid & 31;
    int half = lane >> 4;                        // 0: K0/K1 slice, 1: K2/K3 slice
    int l16  = lane & 15;
    int colBase = wave * 16;

    v8f acc0 = {}, acc1 = {};
    #pragma unroll 4
    for (int kt = 0; kt < DIM / 4; ++kt) {
        int k0 = kt * 4 + half * 2;
        v2f a0, a1, b;
        // B 4x16: row striped across lanes; lanes0-15 rows K0/K1, lanes16-31 rows K2/K3
        b.x = W2[(size_t)k0 * DIM + colBase + l16];
        b.y = W2[(size_t)(k0 + 1) * DIM + colBase + l16];
        // A 16x4 fragments for the two row-tiles
        a0.x = As[l16][k0];       a0.y = As[l16][k0 + 1];
        a1.x = As[16 + l16][k0];  a1.y = As[16 + l16][k0 + 1];
        acc0 = __builtin_amdgcn_wmma_f32_16x16x4_f32(false, a0, false, b,
                                                     (short)0, acc0, false, false);
        acc1 = __builtin_amdgcn_wmma_f32_16x16x4_f32(false, a1, false, b,
                                                     (short)0, acc1, false, false);
    }
    // C/D layout: VGPR j -> lanes0-15: (M=j, N=lane); lanes16-31: (M=8+j, N=lane-16)
    int col = colBase + l16;
    if (r0 + 32 <= n) {                          // fast path: every block but the tail
        #pragma unroll
        for (int j = 0; j < 8; ++j) {
            h2[(size_t)(r0      + j + half * 8) * DIM + col] = acc0[j];
            h2[(size_t)(r0 + 16 + j + half * 8) * DIM + col] = acc1[j];
        }
    } else {
        for (int j = 0; j < 8; ++j) {
            int row0 = r0 + j + half * 8;
            int row1 = row0 + 16;
            if (row0 < n) h2[(size_t)row0 * DIM + col] = acc0[j];
            if (row1 < n) h2[(size_t)row1 * DIM + col] = acc1[j];
        }
    }
}

// ---------- conv2 self-loop seed: agg2[i] = dinv[i]^2 * h2[i] ----------
__global__ __launch_bounds__(256) void k_self2(const float* __restrict__ dinv,
                                               const float* __restrict__ h2,
                                               float* __restrict__ agg, int n) {
    int t = blockIdx.x * 256 + threadIdx.x;
    int i = t >> 5;
    if (i >= n) return;
    int f = (t & 31) * 4;
    float w = dinv[i] * dinv[i];
    const float4 h = *(const float4*)(h2 + (size_t)i * DIM + f);
    float4 o = make_float4(w * h.x, w * h.y, w * h.z, w * h.w);
    *(float4*)(agg + (size_t)i * DIM + f) = o;
}

// ---------- conv2 edge scatter ----------
__global__ __launch_bounds__(256) void k_edge2(const int* __restrict__ src,
                                               const int* __restrict__ dst,
                                               const float* __restrict__ dinv,
                                               const float* __restrict__ h2,
                                               float* __restrict__ agg, int nE) {
    int w = (blockIdx.x * 256 + threadIdx.x) >> 5;
    int lane = threadIdx.x & 31;
    if (w >= nE) return;
    int s = src[w], d = dst[w];
    float nrm = dinv[s] * dinv[d];
    const float* __restrict__ hrow = h2 + (size_t)s * DIM;
    float* __restrict__ orow = agg + (size_t)d * DIM;
    #pragma unroll
    for (int j = 0; j < 4; ++j) {
        int f = lane + 32 * j;
        atomicAdd(&orow[f], nrm * hrow[f]);
    }
}

// ---------- mean pool ----------
__global__ __launch_bounds__(256) void k_zero_pool(float* __restrict__ sums,
                                                   float* __restrict__ cnt, int g) {
    int i = blockIdx.x * 256 + threadIdx.x;
    if (i < g * DIM) sums[i] = 0.f;
    if (i < g) cnt[i] = 0.f;
}
__global__ __launch_bounds__(256) void k_pool(const float* __restrict__ agg2,
                                              const float* __restrict__ b2,
                                              const int* __restrict__ batch,
                                              float* __restrict__ sums,
                                              float* __restrict__ cnt, int n) {
    int w = (blockIdx.x * 256 + threadIdx.x) >> 5;
    int lane = threadIdx.x & 31;
    if (w >= n) return;
    int g = batch[w];
    const float* __restrict__ arow = agg2 + (size_t)w * DIM;
    float* __restrict__ srow = sums + (size_t)g * DIM;
    #pragma unroll
    for (int j = 0; j < 4; ++j) {
        int f = lane + 32 * j;
        atomicAdd(&srow[f], arow[f] + b2[f]);   // fold conv2 bias into pooled sum
    }
    if (lane == 0) atomicAdd(&cnt[g], 1.0f);
}
__global__ __launch_bounds__(256) void k_final(const float* __restrict__ sums,
                                               const float* __restrict__ cnt,
                                               float* __restrict__ out, int g) {
    int i = blockIdx.x * 256 + threadIdx.x;
    if (i >= g * DIM) return;
    float c = cnt[i >> 7];
    out[i] = sums[i] / fmaxf(c, 1.0f);
}

extern "C" void kernel_launch(void* const* d_in, const int* in_sizes, int n_in,
                              void* d_out, int out_size, void* d_ws, size_t ws_size,
                              hipStream_t stream) {
    const int*   node_ids = (const int*)d_in[0];
    const int*   ei       = (const int*)d_in[1];     // [2, E] row-major
    const int*   batch    = (const int*)d_in[2];
    const float* emb      = (const float*)d_in[4];
    const float* W1       = (const float*)d_in[5];
    const float* b1       = (const float*)d_in[6];
    const float* W2       = (const float*)d_in[7];
    const float* b2       = (const float*)d_in[8];
    float* out = (float*)d_out;

    const int N = in_sizes[0];
    const int E = in_sizes[1] / 2;
    const int G = out_size / DIM;
    const int* src = ei;
    const int* dst = ei + E;

    // workspace carve-out (256B aligned)
    char* base = (char*)d_ws;
    size_t off = 0;
    auto alloc = [&](size_t bytes) -> char* {
        char* p = base + off;
        off = (off + bytes + 255) & ~(size_t)255;
        return p;
    };
    float* HW1  = (float*)alloc((size_t)NT * DIM * sizeof(float));   // emb @ W1 table
    float* dinv = (float*)alloc((size_t)N * sizeof(float));          // deg -> rsqrt in place
    float* bufA = (float*)alloc((size_t)N * DIM * sizeof(float));    // agg1, reused as agg2
    float* bufB = (float*)alloc((size_t)N * DIM * sizeof(float));    // h2
    float* sums = (float*)alloc((size_t)G * DIM * sizeof(float));
    float* cnt  = (float*)alloc((size_t)G * sizeof(float));
    (void)ws_size; (void)n_in;

    const int TB = 256;
    dim3 blk(TB);

    // 1) HW1 = emb @ W1
    k_hw1<<<dim3((NT * DIM + TB - 1) / TB), blk, 0, stream>>>(emb, W1, HW1);

    // 2) degree -> dinv
    k_deg_init <<<dim3((N + TB - 1) / TB), blk, 0, stream>>>(dinv, N);
    k_deg_edges<<<dim3((E + TB - 1) / TB), blk, 0, stream>>>(dst, dinv, E);
    k_dinv     <<<dim3((N + TB - 1) / TB), blk, 0, stream>>>(dinv, N);

    // 3) conv1 aggregation (self-loop seed + edge atomics)
    k_self1<<<dim3(((size_t)N * 32 + TB - 1) / TB), blk, 0, stream>>>(node_ids, dinv, HW1, bufA, N);
    k_edge1<<<dim3(((size_t)E * 32 + TB - 1) / TB), blk, 0, stream>>>(src, dst, node_ids, dinv, HW1, bufA, E);

    // 4) h2 = relu(agg1 + b1) @ W2  (WMMA f32 16x16x4, 32-row strips)
    k_gemm2<<<dim3((N + 31) / 32), blk, 0, stream>>>(bufA, b1, W2, bufB, N);

    // 5) conv2 aggregation (overwrites bufA)
    k_self2<<<dim3(((size_t)N * 32 + TB - 1) / TB), blk, 0, stream>>>(dinv, bufB, bufA, N);
    k_edge2<<<dim3(((size_t)E * 32 + TB - 1) / TB), blk, 0, stream>>>(src, dst, dinv, bufB, bufA, E);

    // 6) mean pool (+b2) and finalize
    k_zero_pool<<<dim3((G * DIM + TB - 1) / TB), blk, 0, stream>>>(sums, cnt, G);
    k_pool     <<<dim3(((size_t)N * 32 + TB - 1) / TB), blk, 0, stream>>>(bufA, b2, batch, sums, cnt, N);
    k_final    <<<dim3((G * DIM + TB - 1) / TB), blk, 0, stream>>>(sums, cnt, out, G);
}